// GNN_roman_77764677861851
// MI455X (gfx1250) — compile-verified
//
#include <hip/hip_runtime.h>
#include <hip/hip_bf16.h>
#include <math.h>

typedef float v2f_t __attribute__((ext_vector_type(2)));
typedef float v8f_t __attribute__((ext_vector_type(8)));

#define IN_DIM  128
#define HID_DIM 256
#define H2_DIM  128
#define OUT_DIM 10
#define BN_EPS  1e-5f

// ---------------------------------------------------------------------------
// zero fill (grid-stride)
// ---------------------------------------------------------------------------
__global__ void k_zero(float* __restrict__ p, long n) {
    long i = (long)blockIdx.x * blockDim.x + threadIdx.x;
    long stride = (long)gridDim.x * blockDim.x;
    for (; i < n; i += stride) p[i] = 0.0f;
}

// ---------------------------------------------------------------------------
// degree count: outdeg[src[e]] += 1, indeg[dst[e]] += 1
// ---------------------------------------------------------------------------
__global__ void k_deg(const int* __restrict__ src, const int* __restrict__ dst,
                      float* __restrict__ outdeg, float* __restrict__ indeg, int E) {
    int e = blockIdx.x * blockDim.x + threadIdx.x;
    if (e < E) {
        atomicAdd(&outdeg[src[e]], 1.0f);
        atomicAdd(&indeg[dst[e]], 1.0f);
    }
}

// deg -> rsqrt(max(deg,1)) in place (applied to a contiguous region)
__global__ void k_finalize_dis(float* __restrict__ p, long n) {
    long i = (long)blockIdx.x * blockDim.x + threadIdx.x;
    long stride = (long)gridDim.x * blockDim.x;
    for (; i < n; i += stride) p[i] = rsqrtf(fmaxf(p[i], 1.0f));
}

// ---------------------------------------------------------------------------
// edge scatter: agg[dst[e]][:] += x[src[e]][:] * out_dis[src[e]]
// one float4 chunk per thread; dshift = log2(d/4)
// ---------------------------------------------------------------------------
__global__ void k_scatter(const float* __restrict__ x,
                          const int* __restrict__ src, const int* __restrict__ dst,
                          const float* __restrict__ out_dis,
                          float* __restrict__ agg,
                          long total, int dshift, int d) {
    long idx = (long)blockIdx.x * blockDim.x + threadIdx.x;
    if (idx >= total) return;
    int e = (int)(idx >> dshift);
    int c = (int)(idx & ((1 << dshift) - 1));
    int s = src[e], t = dst[e];
    float sc = out_dis[s];
    const float4* xs = (const float4*)(x + (long)s * d);
    float4 v = xs[c];
    float* ap = agg + (long)t * d + (c << 2);
    atomicAdd(ap + 0, v.x * sc);
    atomicAdd(ap + 1, v.y * sc);
    atomicAdd(ap + 2, v.z * sc);
    atomicAdd(ap + 3, v.w * sc);
}

// ---------------------------------------------------------------------------
// WMMA FP32 GEMM: D = act( (A * scale[:,None]) @ W + b )
//   A: [rows x K], W: [K x cols], scale may be nullptr.
//   One wave per 16x16 output tile; K stepped by 4 with V_WMMA_F32_16X16X4_F32.
//   rows, cols, K all multiples of 16 (guaranteed by the problem dims).
// A fragment (16x4 f32): lane m = lane&15, K = j + 2*(lane>>4)  (contig pair)
// C/D fragment: vgpr i -> (M = i + 8*(lane>>4), N = lane&15)
// ---------------------------------------------------------------------------
__global__ void k_gemm_wmma(const float* __restrict__ A,
                            const float* __restrict__ scale,
                            const float* __restrict__ W,
                            const float* __restrict__ bias,
                            float* __restrict__ D,
                            int rows, int K, int cols, int do_relu) {
    int lane  = threadIdx.x & 31;
    int waveB = threadIdx.x >> 5;
    int wavesPerBlock = blockDim.x >> 5;
    long wave = (long)blockIdx.x * wavesPerBlock + waveB;
    int colTiles = cols >> 4;
    long nTiles = (long)(rows >> 4) * colTiles;
    if (wave >= nTiles) return;          // wave-uniform: EXEC stays all-ones

    int rowTile = (int)(wave / colTiles);
    int colTile = (int)(wave % colTiles);
    int m     = lane & 15;
    int khalf = lane >> 4;               // 0 or 1

    int rowA = (rowTile << 4) + m;       // A row this lane feeds
    int colB = (colTile << 4) + m;       // B col this lane feeds
    float s = (scale != nullptr) ? scale[rowA] : 1.0f;

    const float* Arow = A + (long)rowA * K;

    v8f_t acc = {0.f, 0.f, 0.f, 0.f, 0.f, 0.f, 0.f, 0.f};
    for (int k0 = 0; k0 < K; k0 += 4) {
        int ka = k0 + (khalf << 1);      // this lane's contiguous K pair
        v2f_t a, b;
        a.x = Arow[ka]     * s;
        a.y = Arow[ka + 1] * s;
        b.x = W[(long)ka       * cols + colB];
        b.y = W[(long)(ka + 1) * cols + colB];
        acc = __builtin_amdgcn_wmma_f32_16x16x4_f32(
                  /*neg_a=*/false, a, /*neg_b=*/false, b,
                  /*c_mod=*/(short)0, acc, /*reuse_a=*/false, /*reuse_b=*/false);
    }

    int outCol = (colTile << 4) + (lane & 15);
    float bv = bias[outCol];
    #pragma unroll
    for (int i = 0; i < 8; ++i) {
        int outRow = (rowTile << 4) + i + (khalf << 3);
        float v = acc[i] + bv;
        if (do_relu) v = fmaxf(v, 0.0f);
        D[(long)outRow * cols + outCol] = v;
    }
}

// ---------------------------------------------------------------------------
// sum pooling: emb[graph_ids[n]][f] += x[n][f]   (d = 128, fshift = 7)
// ---------------------------------------------------------------------------
__global__ void k_pool(const float* __restrict__ x, const int* __restrict__ gid,
                       float* __restrict__ emb, long total) {
    long idx = (long)blockIdx.x * blockDim.x + threadIdx.x;
    if (idx >= total) return;
    int n = (int)(idx >> 7);
    int f = (int)(idx & 127);
    atomicAdd(&emb[(long)gid[n] * H2_DIM + f], x[idx]);
}

// ---------------------------------------------------------------------------
// BatchNorm1d (training stats, biased variance). One block of H2 threads.
// ---------------------------------------------------------------------------
__global__ void k_bn(const float* __restrict__ emb,
                     const float* __restrict__ gamma, const float* __restrict__ beta,
                     float* __restrict__ bn, int G) {
    int j = threadIdx.x;                 // column 0..H2-1
    float sum = 0.f, sumsq = 0.f;
    for (int g = 0; g < G; ++g) {
        float v = emb[(long)g * H2_DIM + j];
        sum += v; sumsq += v * v;
    }
    float mean = sum / (float)G;
    float var  = sumsq / (float)G - mean * mean;
    float inv  = rsqrtf(var + BN_EPS);
    float ga = gamma[j], be = beta[j];
    for (int g = 0; g < G; ++g) {
        float v = emb[(long)g * H2_DIM + j];
        bn[(long)g * H2_DIM + j] = (v - mean) * inv * ga + be;
    }
}

// ---------------------------------------------------------------------------
// head: logits = h @ Wf2 + bf2 ; log_softmax. One thread per graph row.
// ---------------------------------------------------------------------------
__global__ void k_head(const float* __restrict__ h, const float* __restrict__ Wf2,
                       const float* __restrict__ bf2, float* __restrict__ outp, int G) {
    int g = blockIdx.x * blockDim.x + threadIdx.x;
    if (g >= G) return;
    float logits[OUT_DIM];
    #pragma unroll
    for (int c = 0; c < OUT_DIM; ++c) logits[c] = bf2[c];
    const float* hr = h + (long)g * H2_DIM;
    for (int k = 0; k < H2_DIM; ++k) {
        float hv = hr[k];
        #pragma unroll
        for (int c = 0; c < OUT_DIM; ++c) logits[c] += hv * Wf2[(long)k * OUT_DIM + c];
    }
    float mx = logits[0];
    #pragma unroll
    for (int c = 1; c < OUT_DIM; ++c) mx = fmaxf(mx, logits[c]);
    float ssum = 0.f;
    #pragma unroll
    for (int c = 0; c < OUT_DIM; ++c) ssum += expf(logits[c] - mx);
    float lse = mx + logf(ssum);
    #pragma unroll
    for (int c = 0; c < OUT_DIM; ++c) outp[(long)g * OUT_DIM + c] = logits[c] - lse;
}

// ---------------------------------------------------------------------------
// launch
// ---------------------------------------------------------------------------
extern "C" void kernel_launch(void* const* d_in, const int* in_sizes, int n_in,
                              void* d_out, int out_size, void* d_ws, size_t ws_size,
                              hipStream_t stream) {
    const float* n_feat = (const float*)d_in[0];
    const int*   src    = (const int*)  d_in[1];
    const int*   dst    = (const int*)  d_in[2];
    const int*   gid    = (const int*)  d_in[3];
    const float* W1  = (const float*)d_in[4];   const float* b1  = (const float*)d_in[5];
    const float* W2  = (const float*)d_in[6];   const float* b2  = (const float*)d_in[7];
    const float* W3  = (const float*)d_in[8];   const float* b3  = (const float*)d_in[9];
    const float* Wf1 = (const float*)d_in[10];  const float* bf1 = (const float*)d_in[11];
    const float* Wf2 = (const float*)d_in[12];  const float* bf2 = (const float*)d_in[13];
    const float* gamma = (const float*)d_in[14];
    const float* beta  = (const float*)d_in[15];

    const int N = in_sizes[0] / IN_DIM;           // 50000
    const int E = in_sizes[1];                    // 800000
    const int G = out_size / (H2_DIM + OUT_DIM);  // 128

    // workspace layout (floats)
    const long Npad = ((long)N + 63) & ~63L;
    float* ws      = (float*)d_ws;
    float* out_dis = ws;                          // [Npad]
    float* in_dis  = out_dis + Npad;              // [Npad]
    float* xbuf    = in_dis  + Npad;              // [N * HID]
    float* aggbuf  = xbuf + (long)N * HID_DIM;    // [N * HID]
    float* bnbuf   = aggbuf + (long)N * HID_DIM;  // [G * H2]
    float* hbuf    = bnbuf + (long)G * H2_DIM;    // [G * H2]

    float* emb_out = (float*)d_out;               // [G * H2]
    float* lp_out  = emb_out + (long)G * H2_DIM;  // [G * OUT]

    const int ZB = 2048, T = 256;

    // --- degrees ---
    k_zero<<<ZB, T, 0, stream>>>(out_dis, 2 * Npad);
    k_deg<<<(E + T - 1) / T, T, 0, stream>>>(src, dst, out_dis, in_dis, E);
    k_finalize_dis<<<ZB, T, 0, stream>>>(out_dis, 2 * Npad);

    // --- layer 1: d_in = 128 -> d_out = 256 ---
    {
        long aggN = (long)N * IN_DIM;
        k_zero<<<ZB, T, 0, stream>>>(aggbuf, aggN);
        long tot = (long)E * (IN_DIM / 4);
        k_scatter<<<(int)((tot + T - 1) / T), T, 0, stream>>>(
            n_feat, src, dst, out_dis, aggbuf, tot, /*dshift=*/5, IN_DIM);
        long waves = (long)(N / 16) * (HID_DIM / 16);
        k_gemm_wmma<<<(int)((waves * 32 + 127) / 128), 128, 0, stream>>>(
            aggbuf, in_dis, W1, b1, xbuf, N, IN_DIM, HID_DIM, 1);
    }
    // --- layer 2: 256 -> 128 ---
    {
        long aggN = (long)N * HID_DIM;
        k_zero<<<ZB, T, 0, stream>>>(aggbuf, aggN);
        long tot = (long)E * (HID_DIM / 4);
        k_scatter<<<(int)((tot + T - 1) / T), T, 0, stream>>>(
            xbuf, src, dst, out_dis, aggbuf, tot, /*dshift=*/6, HID_DIM);
        long waves = (long)(N / 16) * (H2_DIM / 16);
        k_gemm_wmma<<<(int)((waves * 32 + 127) / 128), 128, 0, stream>>>(
            aggbuf, in_dis, W2, b2, xbuf, N, HID_DIM, H2_DIM, 1);
    }
    // --- layer 3: 128 -> 128 ---
    {
        long aggN = (long)N * H2_DIM;
        k_zero<<<ZB, T, 0, stream>>>(aggbuf, aggN);
        long tot = (long)E * (H2_DIM / 4);
        k_scatter<<<(int)((tot + T - 1) / T), T, 0, stream>>>(
            xbuf, src, dst, out_dis, aggbuf, tot, /*dshift=*/5, H2_DIM);
        long waves = (long)(N / 16) * (H2_DIM / 16);
        k_gemm_wmma<<<(int)((waves * 32 + 127) / 128), 128, 0, stream>>>(
            aggbuf, in_dis, W3, b3, xbuf, N, H2_DIM, H2_DIM, 1);
    }

    // --- pooling into d_out (embedding output = pooled sums) ---
    k_zero<<<ZB, T, 0, stream>>>(emb_out, (long)G * H2_DIM);
    {
        long tot = (long)N * H2_DIM;
        k_pool<<<(int)((tot + T - 1) / T), T, 0, stream>>>(xbuf, gid, emb_out, tot);
    }

    // --- BN + FC + head ---
    k_bn<<<1, H2_DIM, 0, stream>>>(emb_out, gamma, beta, bnbuf, G);
    {
        long waves = (long)(G / 16) * (H2_DIM / 16);
        k_gemm_wmma<<<(int)((waves * 32 + 127) / 128), 128, 0, stream>>>(
            bnbuf, nullptr, Wf1, bf1, hbuf, G, H2_DIM, H2_DIM, 1);
    }
    k_head<<<(G + 127) / 128, 128, 0, stream>>>(hbuf, Wf2, bf2, lp_out, G);
}